// DetectionLoss_85186381349371
// MI455X (gfx1250) — compile-verified
//
#include <hip/hip_runtime.h>
#include <hip/hip_bf16.h>

typedef __attribute__((ext_vector_type(2))) float v2f;
typedef __attribute__((ext_vector_type(8))) float v8f;

#define BATCH 64
#define NA_TOT 49392          // 3*(112^2 + 56^2 + 28^2)
#define CHUNK 4096
#define MAXCH 10
#define NSEG (BATCH * 3)

// ---------------------------------------------------------------------------
// Pass 1: per-anchor work. One workgroup per (image, scale-chunk).
// Emits: neg_vals[b*NA_TOT + soff + j] = (neg ? bce : -1)
//        partials[(seg*10+chunk)*5] = {pos_bce, ce, sl1, npos, navail}
// ---------------------------------------------------------------------------
__global__ __launch_bounds__(256) void det_pass1(
    const float* __restrict__ p, const float* __restrict__ anchors,
    const float* __restrict__ tboxes, const int* __restrict__ tlabels,
    float* __restrict__ neg_vals, float* __restrict__ partials,
    int HH, int Na, int nch, int sidx, int soff)
{
    int b = blockIdx.x / nch;
    int chunk = blockIdx.x % nch;
    int seg = b * 3 + sidx;
    int tid = threadIdx.x;

    __shared__ float tbs[64];
    __shared__ int   tls[16];
    __shared__ float red[8][5];
    if (tid < 64) tbs[tid] = tboxes[b * 64 + tid] * (1.0f / 224.0f);
    if (tid < 16) tls[tid] = tlabels[b * 16 + tid];
    __syncthreads();

    float pb = 0.f, ces = 0.f, los = 0.f, npf = 0.f, nvf = 0.f;
    int jend = min(Na, (chunk + 1) * CHUNK);
    for (int j = chunk * CHUNK + tid; j < jend; j += 256) {
        float ax1 = anchors[j * 4 + 0] * (1.0f / 224.0f);
        float ay1 = anchors[j * 4 + 1] * (1.0f / 224.0f);
        float ax2 = anchors[j * 4 + 2] * (1.0f / 224.0f);
        float ay2 = anchors[j * 4 + 3] * (1.0f / 224.0f);
        float aw = ax2 - ax1, ah = ay2 - ay1;
        float area_a = aw * ah;

        float best = -1.0f; int bi = 0;
#pragma unroll
        for (int g = 0; g < 16; ++g) {
            float bx1 = tbs[4 * g + 0], by1 = tbs[4 * g + 1];
            float bx2 = tbs[4 * g + 2], by2 = tbs[4 * g + 3];
            float ix = fmaxf(fminf(ax2, bx2) - fmaxf(ax1, bx1), 0.f);
            float iy = fmaxf(fminf(ay2, by2) - fmaxf(ay1, by1), 0.f);
            float inter = ix * iy;
            float area_b = (bx2 - bx1) * (by2 - by1);
            float iou = inter / (area_a + area_b - inter + 1e-9f);
            if (iou > best) { best = iou; bi = g; }
        }
        bool pos = best >= 0.5f;
        bool neg = best < 0.4f;

        int a = j / HH;
        int pix = j - a * HH;
        size_t base = ((size_t)(b * 3 + a) * 8) * (size_t)HH + (size_t)pix;

        float x = p[base + 4 * (size_t)HH];
        float bce = fmaxf(x, 0.f) - (pos ? x : 0.f) + log1pf(expf(-fabsf(x)));

        if (pos) {
            npf += 1.f; pb += bce;
            // class CE (log_softmax)
            float c0 = p[base + 5 * (size_t)HH];
            float c1 = p[base + 6 * (size_t)HH];
            float c2 = p[base + 7 * (size_t)HH];
            float m = fmaxf(c0, fmaxf(c1, c2));
            float lse = m + logf(expf(c0 - m) + expf(c1 - m) + expf(c2 - m));
            int ml = tls[bi] - 1;
            float selv = (ml == 0) ? c0 : ((ml == 1) ? c1 : c2);
            ces += lse - selv;
            // box encoding + smooth L1
            float mx1 = tbs[4 * bi + 0], my1 = tbs[4 * bi + 1];
            float mx2 = tbs[4 * bi + 2], my2 = tbs[4 * bi + 3];
            float gcx = 0.5f * (mx1 + mx2), gcy = 0.5f * (my1 + my2);
            float gw = mx2 - mx1, gh = my2 - my1;
            float acx = 0.5f * (ax1 + ax2), acy = 0.5f * (ay1 + ay2);
            float e0 = (gcx - acx) / aw;
            float e1 = (gcy - acy) / ah;
            float e2 = logf(gw / aw + 1e-6f);
            float e3 = logf(gh / ah + 1e-6f);
            float b0 = p[base + 0 * (size_t)HH];
            float b1 = p[base + 1 * (size_t)HH];
            float b2 = p[base + 2 * (size_t)HH];
            float b3 = p[base + 3 * (size_t)HH];
            float s = 0.f, d, ad;
            d = b0 - e0; ad = fabsf(d); s += (ad < 1.f) ? 0.5f * d * d : ad - 0.5f;
            d = b1 - e1; ad = fabsf(d); s += (ad < 1.f) ? 0.5f * d * d : ad - 0.5f;
            d = b2 - e2; ad = fabsf(d); s += (ad < 1.f) ? 0.5f * d * d : ad - 0.5f;
            d = b3 - e3; ad = fabsf(d); s += (ad < 1.f) ? 0.5f * d * d : ad - 0.5f;
            los += s;
        }
        if (neg) nvf += 1.f;
        neg_vals[(size_t)b * NA_TOT + soff + j] = neg ? bce : -1.0f;
    }

    // deterministic block reduction (wave32 shfl tree, then fixed-order LDS)
#pragma unroll
    for (int off = 16; off; off >>= 1) {
        pb  += __shfl_xor(pb,  off, 32);
        ces += __shfl_xor(ces, off, 32);
        los += __shfl_xor(los, off, 32);
        npf += __shfl_xor(npf, off, 32);
        nvf += __shfl_xor(nvf, off, 32);
    }
    int wid = tid >> 5, ln = tid & 31;
    if (ln == 0) { red[wid][0] = pb; red[wid][1] = ces; red[wid][2] = los; red[wid][3] = npf; red[wid][4] = nvf; }
    __syncthreads();
    if (tid == 0) {
        float s0 = 0, s1 = 0, s2 = 0, s3 = 0, s4 = 0;
        for (int w = 0; w < 8; ++w) { s0 += red[w][0]; s1 += red[w][1]; s2 += red[w][2]; s3 += red[w][3]; s4 += red[w][4]; }
        float* o = &partials[((size_t)seg * MAXCH + chunk) * 5];
        o[0] = s0; o[1] = s1; o[2] = s2; o[3] = s3; o[4] = s4;
    }
}

// ---------------------------------------------------------------------------
// Pass 2: per (image,scale) segment — exact radix-select top-k sum of negative
// BCEs (keys = float bits, valid since bce >= 0), then segment losses.
// ---------------------------------------------------------------------------
__global__ __launch_bounds__(256) void det_pass2(
    const float* __restrict__ neg_vals, const float* __restrict__ partials,
    float* __restrict__ seg_out)
{
    const int NaTab[3]   = {37632, 9408, 2352};
    const int SoffTab[3] = {0, 37632, 47040};
    const int NchTab[3]  = {10, 3, 1};

    int seg = blockIdx.x;
    int b = seg / 3, s = seg % 3;
    int Na = NaTab[s], soff = SoffTab[s], nch = NchTab[s];
    const float* vals = neg_vals + (size_t)b * NA_TOT + soff;
    int tid = threadIdx.x;

    __shared__ float sh[5];
    __shared__ unsigned hist[256];
    __shared__ unsigned sh_prefix;
    __shared__ int sh_rem;
    __shared__ float swred[8];

    if (tid == 0) {
        float s0 = 0, s1 = 0, s2 = 0, s3 = 0, s4 = 0;
        for (int c = 0; c < nch; ++c) {
            const float* pp = &partials[((size_t)seg * MAXCH + c) * 5];
            s0 += pp[0]; s1 += pp[1]; s2 += pp[2]; s3 += pp[3]; s4 += pp[4];
        }
        sh[0] = s0; sh[1] = s1; sh[2] = s2; sh[3] = s3; sh[4] = s4;
        sh_prefix = 0u;
    }
    __syncthreads();
    float pb = sh[0], ces = sh[1], los = sh[2];
    int npos = (int)(sh[3] + 0.5f);
    int navail = (int)(sh[4] + 0.5f);
    int k = (3 * npos < navail) ? 3 * npos : navail;

    float topk = 0.f;
    if (k > 0) {                           // uniform branch (shared-derived)
        if (tid == 0) sh_rem = k;
        __syncthreads();
        for (int shift = 24; shift >= 0; shift -= 8) {
            hist[tid] = 0u;
            __syncthreads();
            unsigned prefix = sh_prefix;
            unsigned pm = (shift == 24) ? 0u : (0xFFFFFFFFu << (shift + 8));
            for (int j = tid; j < Na; j += 256) {
                float v = vals[j];
                if (v >= 0.f) {
                    unsigned key = __float_as_uint(v);
                    if ((key & pm) == (prefix & pm))
                        atomicAdd(&hist[(key >> shift) & 255u], 1u);
                }
            }
            __syncthreads();
            if (tid == 0) {
                int rem = sh_rem;
                unsigned cum = 0; int bsel = 0;
                for (int bin = 255; bin >= 0; --bin) {
                    unsigned h = hist[bin];
                    if (cum + h >= (unsigned)rem) { bsel = bin; break; }
                    cum += h;
                }
                sh_prefix = prefix | ((unsigned)bsel << shift);
                sh_rem = rem - (int)cum;
            }
            __syncthreads();
        }
        unsigned T = sh_prefix;
        int rem = sh_rem;                  // 1..count(key==T) entries at threshold
        float local = 0.f;                 // deterministic: fixed index order per thread
        for (int j = tid; j < Na; j += 256) {
            float v = vals[j];
            if (v >= 0.f && __float_as_uint(v) > T) local += v;
        }
#pragma unroll
        for (int off = 16; off; off >>= 1) local += __shfl_xor(local, off, 32);
        if ((tid & 31) == 0) swred[tid >> 5] = local;
        __syncthreads();
        if (tid == 0) {
            float sg = 0.f;
            for (int w = 0; w < 8; ++w) sg += swred[w];
            topk = sg + (float)rem * __uint_as_float(T);   // ties are bit-identical: exact
        }
    }

    if (tid == 0) {
        float obj_c = 0.f, cls_c = 0.f, loc_c = 0.f;
        if (npos > 0) {
            float nf = (float)npos;
            cls_c = ces / nf;
            loc_c = los / (4.0f * nf);
            if (navail > 0)
                obj_c = pb / nf + ((k > 0) ? (topk / (float)k) : 0.f);
        }
        seg_out[seg * 3 + 0] = obj_c;
        seg_out[seg * 3 + 1] = cls_c;
        seg_out[seg * 3 + 2] = loc_c;
    }
}

// ---------------------------------------------------------------------------
// Pass 3: single-wave final reduction of 192 segments x 3 quantities using
// V_WMMA_F32_16X16X4_F32 with an all-ones B matrix (D accumulates row sums;
// total = sum over all 64 A slots per WMMA, mapping-invariant).
// ---------------------------------------------------------------------------
__global__ __launch_bounds__(32) void det_final(
    const float* __restrict__ seg_out, float* __restrict__ out)
{
    int lane = threadIdx.x;               // wave32, EXEC all ones
    v2f ones; ones.x = 1.0f; ones.y = 1.0f;
    float tot[3];
#pragma unroll
    for (int q = 0; q < 3; ++q) {
        v8f c = {};
#pragma unroll
        for (int t = 0; t < 3; ++t) {     // 3 * 64 = 192 segments
            v2f a;
            a.x = seg_out[(t * 64 + lane * 2 + 0) * 3 + q];
            a.y = seg_out[(t * 64 + lane * 2 + 1) * 3 + q];
            c = __builtin_amdgcn_wmma_f32_16x16x4_f32(
                    false, a, false, ones, (short)0, c, false, false);
        }
        // lanes l and l+16 share column n=l; sum 8 VGPRs then fold halves
        float t8 = c[0] + c[1] + c[2] + c[3] + c[4] + c[5] + c[6] + c[7];
        t8 += __shfl_xor(t8, 16, 32);
        tot[q] = t8;
    }
    if (lane == 0) {
        float o = tot[0] * (1.0f / BATCH);
        float cl = tot[1] * (1.0f / BATCH);
        float lo = tot[2] * (1.0f / BATCH);
        out[0] = o; out[1] = cl; out[2] = lo; out[3] = o + cl + lo;
    }
}

// ---------------------------------------------------------------------------
extern "C" void kernel_launch(void* const* d_in, const int* in_sizes, int n_in,
                              void* d_out, int out_size, void* d_ws, size_t ws_size,
                              hipStream_t stream) {
    const float* p0  = (const float*)d_in[0];
    const float* p1  = (const float*)d_in[1];
    const float* p2  = (const float*)d_in[2];
    const float* an0 = (const float*)d_in[3];
    const float* an1 = (const float*)d_in[4];
    const float* an2 = (const float*)d_in[5];
    const float* tb  = (const float*)d_in[6];
    const int*   tl  = (const int*)d_in[7];

    float* ws       = (float*)d_ws;
    float* neg_vals = ws;                                    // 64*49392 floats
    float* partials = ws + (size_t)BATCH * NA_TOT;           // 192*10*5 floats
    float* seg_out  = partials + (size_t)NSEG * MAXCH * 5;   // 192*3 floats

    det_pass1<<<BATCH * 10, 256, 0, stream>>>(p0, an0, tb, tl, neg_vals, partials,
                                              112 * 112, 37632, 10, 0, 0);
    det_pass1<<<BATCH * 3, 256, 0, stream>>>(p1, an1, tb, tl, neg_vals, partials,
                                             56 * 56, 9408, 3, 1, 37632);
    det_pass1<<<BATCH * 1, 256, 0, stream>>>(p2, an2, tb, tl, neg_vals, partials,
                                             28 * 28, 2352, 1, 2, 47040);
    det_pass2<<<NSEG, 256, 0, stream>>>(neg_vals, partials, seg_out);
    det_final<<<1, 32, 0, stream>>>(seg_out, (float*)d_out);
}